// SAMS_NET_59596966199343
// MI455X (gfx1250) — compile-verified
//
#include <hip/hip_runtime.h>

// ============================================================================
// SAMS-Net forward on MI455X (gfx1250, wave32).
//  - All 1x1 convs: fp32 WMMA GEMM (v_wmma_f32_16x16x4_f32). One wave computes
//    a 16(outCh) x 64(points) strip: A fragment loaded once per k-step, reused
//    across 4 N-tiles (4 independent accumulators -> pipelines the XDL and
//    covers WMMA->WMMA hazards with independent work).
//  - Fast path (full M tile, Cin%4==0) has NO per-element guards -> plain
//    global_load_b32 with immediate offsets (N hardcoded so channel stride is
//    a compile-time immediate). Ragged path uses clamped indices + mask
//    multiplies (cndmask) instead of exec-divergent conditional loads.
//  - LSE + attentive-pool fused per point (one wave/point), score GEMM in
//    WMMA, softmax over the 16 neighbors via 16-lane shfl_xor reductions.
//  - KNN brute force with LDS candidate tiling.
// ============================================================================

typedef __attribute__((ext_vector_type(2))) float v2f;
typedef __attribute__((ext_vector_type(8))) float v8f;

#define EPS  1e-6f
#define NPTS 6144
#define NB   2

// ---------------------------------------------------------------------------
// KNN: one thread per query point, candidates tiled through LDS.
// ---------------------------------------------------------------------------
#define KNN_TILE 256
__global__ __launch_bounds__(256) void knn_kernel(const float* __restrict__ pts,
                                                  int* __restrict__ knn_idx,
                                                  float* __restrict__ knn_dist) {
  __shared__ float sx[KNN_TILE], sy[KNN_TILE], sz[KNN_TILE], ssq[KNN_TILE];
  const int tid = threadIdx.x;
  const int q = blockIdx.x * KNN_TILE + tid;   // NPTS % 256 == 0 -> uniform b/block
  const int b = q / NPTS;
  const int n = q - b * NPTS;
  const float* base = pts + (size_t)b * NPTS * 3;
  const float qx = base[n * 3 + 0], qy = base[n * 3 + 1], qz = base[n * 3 + 2];
  const float qsq = qx * qx + qy * qy + qz * qz;

  float bd[16];
  int bi[16];
#pragma unroll
  for (int i = 0; i < 16; i++) { bd[i] = 3.0e38f; bi[i] = 0; }

  for (int m0 = 0; m0 < NPTS; m0 += KNN_TILE) {
    __syncthreads();
    const int m = m0 + tid;
    const float cx = base[m * 3 + 0], cy = base[m * 3 + 1], cz = base[m * 3 + 2];
    sx[tid] = cx; sy[tid] = cy; sz[tid] = cz;
    ssq[tid] = cx * cx + cy * cy + cz * cz;
    __syncthreads();
    for (int j = 0; j < KNN_TILE; j++) {
      const float d2 = qsq + ssq[j] - 2.0f * (qx * sx[j] + qy * sy[j] + qz * sz[j]);
      if (d2 < bd[15]) {
        int pos = 15;
        while (pos > 0 && bd[pos - 1] > d2) {
          bd[pos] = bd[pos - 1]; bi[pos] = bi[pos - 1]; pos--;
        }
        bd[pos] = d2; bi[pos] = m0 + j;
      }
    }
  }
#pragma unroll
  for (int k = 0; k < 16; k++) {
    knn_idx[(size_t)q * 16 + k] = bi[k];
    knn_dist[(size_t)q * 16 + k] = sqrtf(fmaxf(bd[k], 1e-12f));
  }
}

// ---------------------------------------------------------------------------
// x0 = concat(latent^T, points^T): (B, 35, N)
// ---------------------------------------------------------------------------
__global__ void concat_inputs(const float* __restrict__ latent,
                              const float* __restrict__ points,
                              float* __restrict__ out) {
  const int t = blockIdx.x * blockDim.x + threadIdx.x;
  if (t >= NB * 35 * NPTS) return;
  const int n = t % NPTS;
  const int ch = (t / NPTS) % 35;
  const int b = t / (35 * NPTS);
  float v;
  if (ch < 32) v = latent[((size_t)b * NPTS + n) * 32 + ch];
  else         v = points[((size_t)b * NPTS + n) * 3 + (ch - 32)];
  out[t] = v;
}

__global__ void transpose_lat(const float* __restrict__ latent,
                              float* __restrict__ out) {
  const int t = blockIdx.x * blockDim.x + threadIdx.x;
  if (t >= NB * 32 * NPTS) return;
  const int n = t % NPTS;
  const int ch = (t / NPTS) % 32;
  const int b = t / (32 * NPTS);
  out[t] = latent[((size_t)b * NPTS + n) * 32 + ch];
}

// ---------------------------------------------------------------------------
// conv1x1 fp32 WMMA GEMM. grid = (ceil(Cout/16), NPTS/64, B), block = 32.
// act: 0 none, 1 relu, 2 lrelu(0.2), 3 lrelu(0.01).  accum: Y += before act.
// ---------------------------------------------------------------------------
__global__ __launch_bounds__(32) void conv1x1_wmma(
    const float* __restrict__ X, const float* __restrict__ W,
    const float* __restrict__ bias,
    const float* __restrict__ bn_be, const float* __restrict__ bn_g,
    const float* __restrict__ bn_m, const float* __restrict__ bn_v,
    float* __restrict__ Y, int Cin, int Cout, int act, int accum) {
  const int lane = threadIdx.x & 31;
  const int r16 = lane & 15, hi = lane >> 4;
  const int mt = blockIdx.x, bb = blockIdx.z;
  const int col0 = blockIdx.y * 64 + r16;
  const int row = mt * 16 + r16;
  const float* Xb = X + (size_t)bb * Cin * NPTS;

  v8f accs[4] = {v8f{}, v8f{}, v8f{}, v8f{}};

  const bool fullTile = ((mt + 1) * 16 <= Cout) && ((Cin & 3) == 0);
  if (fullTile) {
    // -------- branch-free fast path --------
    const float* wp = W + (size_t)row * Cin + 2 * hi;
    const float* xp = Xb + (size_t)(2 * hi) * NPTS + col0;
    for (int k0 = 0; k0 < Cin; k0 += 4) {
      v2f a;
      a.x = wp[0]; a.y = wp[1];
      wp += 4;
      __builtin_prefetch(xp + 4 * NPTS, 0, 0);
      v2f b0, b1, b2, b3;
      b0.x = xp[0];      b0.y = xp[NPTS];
      b1.x = xp[16];     b1.y = xp[NPTS + 16];
      b2.x = xp[32];     b2.y = xp[NPTS + 32];
      b3.x = xp[48];     b3.y = xp[NPTS + 48];
      xp += 4 * (size_t)NPTS;
      accs[0] = __builtin_amdgcn_wmma_f32_16x16x4_f32(false, a, false, b0, (short)0, accs[0], false, false);
      accs[1] = __builtin_amdgcn_wmma_f32_16x16x4_f32(false, a, false, b1, (short)0, accs[1], false, false);
      accs[2] = __builtin_amdgcn_wmma_f32_16x16x4_f32(false, a, false, b2, (short)0, accs[2], false, false);
      accs[3] = __builtin_amdgcn_wmma_f32_16x16x4_f32(false, a, false, b3, (short)0, accs[3], false, false);
    }
  } else {
    // -------- ragged path: clamped indices + mask multiplies (no branches) --
    const int rowC = (row < Cout) ? row : (Cout - 1);
    const float mrow = (row < Cout) ? 1.0f : 0.0f;
    const float* wr = W + (size_t)rowC * Cin;
    for (int k0 = 0; k0 < Cin; k0 += 4) {
      const int ka = k0 + 2 * hi;
      const float m0 = (ka < Cin) ? 1.0f : 0.0f;
      const float m1 = (ka + 1 < Cin) ? 1.0f : 0.0f;
      const int ka0 = (ka < Cin) ? ka : (Cin - 1);
      const int ka1 = (ka + 1 < Cin) ? (ka + 1) : (Cin - 1);
      v2f a;
      a.x = wr[ka0] * (mrow * m0);
      a.y = wr[ka1] * (mrow * m1);
      const float* x0 = Xb + (size_t)ka0 * NPTS + col0;
      const float* x1 = Xb + (size_t)ka1 * NPTS + col0;
      v2f b0, b1, b2, b3;
      b0.x = x0[0]  * m0; b0.y = x1[0]  * m1;
      b1.x = x0[16] * m0; b1.y = x1[16] * m1;
      b2.x = x0[32] * m0; b2.y = x1[32] * m1;
      b3.x = x0[48] * m0; b3.y = x1[48] * m1;
      accs[0] = __builtin_amdgcn_wmma_f32_16x16x4_f32(false, a, false, b0, (short)0, accs[0], false, false);
      accs[1] = __builtin_amdgcn_wmma_f32_16x16x4_f32(false, a, false, b1, (short)0, accs[1], false, false);
      accs[2] = __builtin_amdgcn_wmma_f32_16x16x4_f32(false, a, false, b2, (short)0, accs[2], false, false);
      accs[3] = __builtin_amdgcn_wmma_f32_16x16x4_f32(false, a, false, b3, (short)0, accs[3], false, false);
    }
  }

  // -------- epilogue: bias/bn/act/store --------
#pragma unroll
  for (int r = 0; r < 8; r++) {
    const int o = mt * 16 + r + 8 * hi;
    if (o >= Cout) continue;
    float bia = bias[o];
    float sc = 1.0f, sh = 0.0f;
    if (bn_g) {
      sc = bn_g[o] * rsqrtf(bn_v[o] + EPS);
      sh = bn_be[o] - bn_m[o] * sc;
    }
    const size_t orow = ((size_t)bb * Cout + o) * NPTS + col0;
#pragma unroll
    for (int t = 0; t < 4; t++) {
      float y = (accs[t][r] + bia) * sc + sh;
      const size_t oi = orow + 16 * t;
      if (accum) y += Y[oi];
      if (act == 1)      y = fmaxf(y, 0.0f);
      else if (act == 2) y = (y >= 0.0f) ? y : 0.2f * y;
      else if (act == 3) y = (y >= 0.0f) ? y : 0.01f * y;
      Y[oi] = y;
    }
  }
}

// ---------------------------------------------------------------------------
// Fused LSE + attentive pool: one wave per point, 4 waves per block.
// ---------------------------------------------------------------------------
#define LSE_WAVES 4
__global__ __launch_bounds__(128) void lse_pool_kernel(
    const float* __restrict__ coords, const int* __restrict__ knn_idx,
    const float* __restrict__ knn_dist,
    const float* __restrict__ feats,                              // (B, h, N)
    const float* __restrict__ Wl, const float* __restrict__ bl,   // (h x 10)
    const float* __restrict__ l_be, const float* __restrict__ l_g,
    const float* __restrict__ l_m, const float* __restrict__ l_v,
    const float* __restrict__ Ws,                                 // (dout x dout)
    const float* __restrict__ Wp, const float* __restrict__ bp,   // (co x dout)
    const float* __restrict__ p_be, const float* __restrict__ p_g,
    const float* __restrict__ p_m, const float* __restrict__ p_v,
    float* __restrict__ out,                                      // (B, co, N)
    int h, int dout, int co) {
  __shared__ float s_xcat[LSE_WAVES][128 * 16];
  __shared__ float s_tile[LSE_WAVES][16 * 16];
  __shared__ float s_feat[LSE_WAVES][128];

  const int lane = threadIdx.x & 31;
  const int wave = threadIdx.x >> 5;
  const int pt = blockIdx.x * LSE_WAVES + wave;
  const int b = pt / NPTS;
  const int n = pt - b * NPTS;
  const int k = lane & 15, hi = lane >> 4;

  float* xcat = s_xcat[wave];
  float* stile = s_tile[wave];
  float* featb = s_feat[wave];

  // relative position encoding inputs for neighbor k
  const int nbr = knn_idx[(size_t)pt * 16 + k];
  const float dk = knn_dist[(size_t)pt * 16 + k];
  const float* cb = coords + (size_t)b * NPTS * 3;
  float cat[10];
  cat[0] = cb[n * 3 + 0]; cat[1] = cb[n * 3 + 1]; cat[2] = cb[n * 3 + 2];
  cat[3] = cb[nbr * 3 + 0]; cat[4] = cb[nbr * 3 + 1]; cat[5] = cb[nbr * 3 + 2];
  cat[6] = cat[0] - cat[3]; cat[7] = cat[1] - cat[4]; cat[8] = cat[2] - cat[5];
  cat[9] = dk;

  // enc channels (half-wave handles odd/even channel) + neighbor gather
  const float* fb = feats + (size_t)b * h * NPTS;
  for (int c0 = 0; c0 < h; c0 += 2) {
    const int ch = c0 + hi;
    float a = bl[ch];
#pragma unroll
    for (int i = 0; i < 10; i++) a += Wl[ch * 10 + i] * cat[i];
    const float sc = l_g[ch] * rsqrtf(l_v[ch] + EPS);
    const float y = (a - l_m[ch]) * sc + l_be[ch];
    xcat[ch * 16 + k] = fmaxf(y, 0.0f);
    xcat[(h + ch) * 16 + k] = fb[(size_t)ch * NPTS + nbr];
  }
  __syncthreads();

  // score GEMM (WMMA) + per-channel softmax over 16 neighbors
  const int mtiles = dout >> 4;
  for (int mt = 0; mt < mtiles; mt++) {
    v8f acc = {};
    const float* wsr = Ws + (size_t)(mt * 16 + k) * dout + 2 * hi;
    const float* xr = xcat + (2 * hi) * 16 + k;
    for (int k0 = 0; k0 < dout; k0 += 4) {
      v2f a, bv;
      a.x = wsr[0]; a.y = wsr[1];
      wsr += 4;
      bv.x = xr[0]; bv.y = xr[16];
      xr += 64;
      acc = __builtin_amdgcn_wmma_f32_16x16x4_f32(false, a, false, bv,
                                                  (short)0, acc, false, false);
    }
#pragma unroll
    for (int r = 0; r < 8; r++) stile[(r + 8 * hi) * 16 + k] = acc[r];
    __syncthreads();

#pragma unroll
    for (int cc = 0; cc < 8; cc++) {
      const int mloc = cc * 2 + hi;
      const int o = mt * 16 + mloc;
      const float v = stile[mloc * 16 + k];
      float mx = v;
      for (int off = 1; off < 16; off <<= 1) mx = fmaxf(mx, __shfl_xor(mx, off, 32));
      const float e = __expf(v - mx);
      float sm = e;
      for (int off = 1; off < 16; off <<= 1) sm += __shfl_xor(sm, off, 32);
      float contrib = (e / sm) * xcat[o * 16 + k];
      for (int off = 1; off < 16; off <<= 1) contrib += __shfl_xor(contrib, off, 32);
      if (k == 0) featb[o] = contrib;
    }
    __syncthreads();
  }

  // pool conv (co x dout) on pooled feature + bn + relu
  for (int c2 = lane; c2 < co; c2 += 32) {
    float a = bp[c2];
    for (int o = 0; o < dout; o++) a += Wp[(size_t)c2 * dout + o] * featb[o];
    const float sc = p_g[c2] * rsqrtf(p_v[c2] + EPS);
    const float y = (a - p_m[c2]) * sc + p_be[c2];
    out[((size_t)b * co + c2) * NPTS + n] = fmaxf(y, 0.0f);
  }
}

// ===========================================================================
extern "C" void kernel_launch(void* const* d_in, const int* in_sizes, int n_in,
                              void* d_out, int out_size, void* d_ws, size_t ws_size,
                              hipStream_t stream) {
  (void)in_sizes; (void)out_size; (void)ws_size;
  const int B = NB, N = NPTS;
  static const int DIN[4]  = {35, 32, 64, 128};
  static const int DOUT[4] = {16, 32, 64, 128};

  const float* latent = (const float*)d_in[0];
  const float* points = (const float*)d_in[1];

  // ---- parameter leaves in JAX tree-flatten order (sorted dict keys) ----
  int lsz[160]; int nl = 0;
  for (int l = 0; l < 4; l++) {
    const int din = DIN[l], dO = DOUT[l], h = dO / 2;
    lsz[nl++] = h; lsz[nl++] = h; lsz[nl++] = h; lsz[nl++] = h; lsz[nl++] = h * 10; lsz[nl++] = h;      // lse1
    lsz[nl++] = h; lsz[nl++] = h; lsz[nl++] = h; lsz[nl++] = h; lsz[nl++] = h * 10; lsz[nl++] = h;      // lse2
    lsz[nl++] = h * din; lsz[nl++] = h;                                                                  // mlp1
    lsz[nl++] = 2 * dO * dO; lsz[nl++] = 2 * dO;                                                         // mlp2
    lsz[nl++] = h; lsz[nl++] = h; lsz[nl++] = h; lsz[nl++] = h; lsz[nl++] = h * dO; lsz[nl++] = h; lsz[nl++] = dO * dO;       // pool1
    lsz[nl++] = dO; lsz[nl++] = dO; lsz[nl++] = dO; lsz[nl++] = dO; lsz[nl++] = dO * dO; lsz[nl++] = dO; lsz[nl++] = dO * dO; // pool2
    lsz[nl++] = 2 * dO; lsz[nl++] = 2 * dO; lsz[nl++] = 2 * dO; lsz[nl++] = 2 * dO; lsz[nl++] = 2 * dO * din; lsz[nl++] = 2 * dO; // shortcut
  }
  lsz[nl++] = 256;     lsz[nl++] = 1;   // sdf {W,b}
  lsz[nl++] = 8 * 256; lsz[nl++] = 8;   // seg {W,b}

  const float* leaf[160];
  if (n_in >= 2 + nl) {
    for (int i = 0; i < nl; i++) leaf[i] = (const float*)d_in[2 + i];
  } else {
    const float* base = (const float*)d_in[2];
    size_t off = 0;
    for (int i = 0; i < nl; i++) { leaf[i] = base + off; off += (size_t)lsz[i]; }
  }

  // ---- workspace layout (~40 MB fp32) ----
  char* ws = (char*)d_ws;
  int*   kidx = (int*)ws;                               // B*N*16
  float* kdst = (float*)(ws + (size_t)196608 * 4);      // B*N*16
  float* bufX = kdst + 196608;                          // 2*256*6144
  float* bufY = bufX + 3145728;
  float* x1b  = bufY + 3145728;                         // 2*64*6144
  float* p1b  = x1b + 786432;
  float* p2b  = p1b + 786432;                           // 2*128*6144

  knn_kernel<<<(B * N) / KNN_TILE, KNN_TILE, 0, stream>>>(points, kidx, kdst);
  concat_inputs<<<(B * 35 * N + 255) / 256, 256, 0, stream>>>(latent, points, bufX);

  float* curX = bufX; float* curY = bufY;
  int c = 0;
  for (int l = 0; l < 4; l++) {
    const int din = DIN[l], dO = DOUT[l], h = dO / 2;
    const float *l1_be = leaf[c], *l1_g = leaf[c+1], *l1_m = leaf[c+2], *l1_v = leaf[c+3], *l1_W = leaf[c+4], *l1_b = leaf[c+5]; c += 6;
    const float *l2_be = leaf[c], *l2_g = leaf[c+1], *l2_m = leaf[c+2], *l2_v = leaf[c+3], *l2_W = leaf[c+4], *l2_b = leaf[c+5]; c += 6;
    const float *m1_W = leaf[c], *m1_b = leaf[c+1]; c += 2;
    const float *m2_W = leaf[c], *m2_b = leaf[c+1]; c += 2;
    const float *q1_be = leaf[c], *q1_g = leaf[c+1], *q1_m = leaf[c+2], *q1_v = leaf[c+3], *q1_W = leaf[c+4], *q1_b = leaf[c+5], *q1_S = leaf[c+6]; c += 7;
    const float *q2_be = leaf[c], *q2_g = leaf[c+1], *q2_m = leaf[c+2], *q2_v = leaf[c+3], *q2_W = leaf[c+4], *q2_b = leaf[c+5], *q2_S = leaf[c+6]; c += 7;
    const float *sc_be = leaf[c], *sc_g = leaf[c+1], *sc_m = leaf[c+2], *sc_v = leaf[c+3], *sc_W = leaf[c+4], *sc_b = leaf[c+5]; c += 6;

    conv1x1_wmma<<<dim3((h + 15) / 16, N / 64, B), 32, 0, stream>>>(
        curX, m1_W, m1_b, nullptr, nullptr, nullptr, nullptr, x1b, din, h, 2, 0);
    lse_pool_kernel<<<(B * N) / LSE_WAVES, 128, 0, stream>>>(
        points, kidx, kdst, x1b, l1_W, l1_b, l1_be, l1_g, l1_m, l1_v,
        q1_S, q1_W, q1_b, q1_be, q1_g, q1_m, q1_v, p1b, h, dO, h);
    lse_pool_kernel<<<(B * N) / LSE_WAVES, 128, 0, stream>>>(
        points, kidx, kdst, p1b, l2_W, l2_b, l2_be, l2_g, l2_m, l2_v,
        q2_S, q2_W, q2_b, q2_be, q2_g, q2_m, q2_v, p2b, h, dO, dO);
    conv1x1_wmma<<<dim3((2 * dO + 15) / 16, N / 64, B), 32, 0, stream>>>(
        curX, sc_W, sc_b, sc_be, sc_g, sc_m, sc_v, curY, din, 2 * dO, 0, 0);
    conv1x1_wmma<<<dim3((2 * dO + 15) / 16, N / 64, B), 32, 0, stream>>>(
        p2b, m2_W, m2_b, nullptr, nullptr, nullptr, nullptr, curY, dO, 2 * dO, 3, 1);

    float* t = curX; curX = curY; curY = t;
  }
  const float *sdf_W = leaf[c], *sdf_b = leaf[c+1], *seg_W = leaf[c+2], *seg_b = leaf[c+3];

  float* of = (float*)d_out;
  transpose_lat<<<(B * 32 * N + 255) / 256, 256, 0, stream>>>(latent, of);
  conv1x1_wmma<<<dim3(1, N / 64, B), 32, 0, stream>>>(
      curX, sdf_W, sdf_b, nullptr, nullptr, nullptr, nullptr,
      of + (size_t)B * 32 * N, 256, 1, 0, 0);
  conv1x1_wmma<<<dim3(1, N / 64, B), 32, 0, stream>>>(
      curX, seg_W, seg_b, nullptr, nullptr, nullptr, nullptr,
      of + (size_t)B * 32 * N + (size_t)B * N, 256, 8, 0, 0);
}